// M7_19164144074973
// MI455X (gfx1250) — compile-verified
//
#include <hip/hip_runtime.h>
#include <hip/hip_bf16.h>

#define NND   50000
#define NED   1600000
#define NET   (NED + NND)      // edges + self loops
#define HC    128
#define FE    16

typedef float v2f __attribute__((ext_vector_type(2)));
typedef float v8f __attribute__((ext_vector_type(8)));

// ---------- helpers ----------
__device__ __forceinline__ void atomAddF(float* p, float v) {
    unsafeAtomicAdd(p, v);   // hardware global_atomic_add_f32
}
__device__ __forceinline__ unsigned encodeF(float f) {
    unsigned u = __float_as_uint(f);
    return (u & 0x80000000u) ? ~u : (u | 0x80000000u);
}
__device__ __forceinline__ float decodeF(unsigned u) {
    return (u & 0x80000000u) ? __uint_as_float(u & 0x7FFFFFFFu) : __uint_as_float(~u);
}
#define AMAX_INIT 0x00800000u   // encodeF(-FLT_MAX)

__global__ void k_zero_f32(float* p, int n) {
    int i = blockIdx.x * blockDim.x + threadIdx.x;
    if (i < n) p[i] = 0.f;
}
__global__ void k_fill_u32(unsigned* p, int n, unsigned v) {
    int i = blockIdx.x * blockDim.x + threadIdx.x;
    if (i < n) p[i] = v;
}

// ---------- self-loop attr (fill_value='mean') ----------
__global__ void k_deg_scatter(const int* __restrict__ ei, const float* __restrict__ ea,
                              float* deg, float* loop_attr) {
    int e = blockIdx.x * blockDim.x + threadIdx.x;
    if (e >= NED) return;
    int dst = ei[NED + e];
    atomAddF(&deg[dst], 1.f);
    const float* a = ea + (size_t)e * FE;
    float* la = loop_attr + (size_t)dst * FE;
    #pragma unroll
    for (int f = 0; f < FE; ++f) atomAddF(&la[f], a[f]);
}
__global__ void k_loop_norm(float* loop_attr, const float* deg) {
    int i = blockIdx.x * blockDim.x + threadIdx.x;
    if (i >= NND * FE) return;
    loop_attr[i] /= fmaxf(deg[i >> 4], 1.f);
}

// ---------- WMMA f32 GEMM:  C[M,128] = A[M,K] @ W[K,128] + bias ----------
// block (32,4): each wave owns one 16x16 tile; grid (M/16, 2)
__global__ void k_gemm_wmma(const float* __restrict__ A, const float* __restrict__ W,
                            const float* __restrict__ bias, float* __restrict__ C, int K) {
    int lane = threadIdx.x;
    int half = lane >> 4;          // K sub-block (0/1) and row sub-block for C
    int l16  = lane & 15;
    int rowTile = blockIdx.x;
    int colTile = blockIdx.y * 4 + threadIdx.y;   // 0..7
    int row = rowTile * 16 + l16;                 // A row held by this lane
    int col = colTile * 16 + l16;                 // B/C column held by this lane
    v8f acc = {};
    for (int k = 0; k < K; k += 4) {
        int ka = k + half * 2;
        v2f a, b;
        a.x = A[(size_t)row * K + ka];
        a.y = A[(size_t)row * K + ka + 1];
        b.x = W[(size_t)ka * HC + col];
        b.y = W[(size_t)(ka + 1) * HC + col];
        acc = __builtin_amdgcn_wmma_f32_16x16x4_f32(false, a, false, b,
                                                    (short)0, acc, false, false);
    }
    float bv = bias[col];
    #pragma unroll
    for (int i = 0; i < 8; ++i) {
        int m = rowTile * 16 + i + 8 * half;
        C[(size_t)m * HC + col] = acc[i] + bv;
    }
}

// ---------- edge pass A: attention logits + segment max ----------
// one wave per edge; lane j owns channels 4j..4j+3 (head h = j>>3)
__global__ void k_edge_a(const float* __restrict__ xl, const float* __restrict__ xr,
                         const int* __restrict__ ei, const float* __restrict__ ea,
                         const float* __restrict__ loop_attr,
                         const float* __restrict__ We, const float* __restrict__ att,
                         float* __restrict__ a_out, unsigned* __restrict__ amax) {
    __shared__ __align__(16) float sWe[FE * HC];
    __shared__ __align__(16) float sAtt[HC];
    int tid = threadIdx.y * 32 + threadIdx.x;
    for (int i = tid; i < FE * HC; i += 256) sWe[i] = We[i];
    if (tid < HC) sAtt[tid] = att[tid];
    __syncthreads();

    int e = blockIdx.x * blockDim.y + threadIdx.y;
    if (e >= NET) return;
    int j = threadIdx.x;        // 0..31
    int h = j >> 3;             // head 0..3
    int src, dst; const float* eav;
    if (e < NED) { src = ei[e]; dst = ei[NED + e]; eav = ea + (size_t)e * FE; }
    else         { src = e - NED; dst = src;       eav = loop_attr + (size_t)src * FE; }

    float myea = (j < FE) ? eav[j] : 0.f;
    const float4* sWe4 = reinterpret_cast<const float4*>(sWe);
    float4 eev = {0.f, 0.f, 0.f, 0.f};
    #pragma unroll
    for (int k = 0; k < FE; ++k) {
        float ev = __shfl(myea, k, 32);
        float4 w = sWe4[k * 32 + j];                  // ds_load_b128
        eev.x += ev * w.x; eev.y += ev * w.y;
        eev.z += ev * w.z; eev.w += ev * w.w;
    }
    float4 xlv  = *reinterpret_cast<const float4*>(xl + (size_t)src * HC + 4 * j);
    float4 xrv  = *reinterpret_cast<const float4*>(xr + (size_t)dst * HC + 4 * j);
    float4 attv = *reinterpret_cast<const float4*>(sAtt + 4 * j);
    float s = 0.f, v;
    v = xlv.x + xrv.x + eev.x; v = (v > 0.f) ? v : 0.2f * v; s += v * attv.x;
    v = xlv.y + xrv.y + eev.y; v = (v > 0.f) ? v : 0.2f * v; s += v * attv.y;
    v = xlv.z + xrv.z + eev.z; v = (v > 0.f) ? v : 0.2f * v; s += v * attv.z;
    v = xlv.w + xrv.w + eev.w; v = (v > 0.f) ? v : 0.2f * v; s += v * attv.w;
    // reduce within 8-lane head group
    s += __shfl_xor(s, 4, 32);
    s += __shfl_xor(s, 2, 32);
    s += __shfl_xor(s, 1, 32);
    if ((j & 7) == 0) {                 // lanes 0,8,16,24: one per head
        a_out[(size_t)e * 4 + h] = s;
        atomicMax(&amax[dst * 4 + h], encodeF(s));
    }
}

// ---------- edge pass B: exp + segment sum ----------
__global__ void k_edge_b(float* __restrict__ aex, const int* __restrict__ ei,
                         const unsigned* __restrict__ amax, float* __restrict__ den) {
    int idx = blockIdx.x * blockDim.x + threadIdx.x;
    if (idx >= NET * 4) return;
    int e = idx >> 2, t = idx & 3;
    int dst = (e < NED) ? ei[NED + e] : (e - NED);
    float ex = __expf(aex[idx] - decodeF(amax[dst * 4 + t]));
    aex[idx] = ex;
    atomAddF(&den[dst * 4 + t], ex);
}

// ---------- edge pass C: weighted aggregation ----------
__global__ void k_edge_c(const float* __restrict__ xl, const int* __restrict__ ei,
                         const float* __restrict__ aex, const float* __restrict__ den,
                         float* __restrict__ outb) {
    int e = blockIdx.x * blockDim.y + threadIdx.y;
    if (e >= NET) return;
    int j = threadIdx.x;
    int h = j >> 3;
    int src, dst;
    if (e < NED) { src = ei[e]; dst = ei[NED + e]; }
    else         { src = e - NED; dst = src; }
    float alpha = aex[(size_t)e * 4 + h] / (den[dst * 4 + h] + 1e-16f);
    float4 xlv = *reinterpret_cast<const float4*>(xl + (size_t)src * HC + 4 * j);
    float* ob = outb + (size_t)dst * HC + 4 * j;
    atomAddF(ob + 0, alpha * xlv.x);
    atomAddF(ob + 1, alpha * xlv.y);
    atomAddF(ob + 2, alpha * xlv.z);
    atomAddF(ob + 3, alpha * xlv.w);
}

// ---------- bias add (float4) ----------
__global__ void k_add_bias(const float4* __restrict__ in, const float4* __restrict__ bias,
                           float4* __restrict__ out) {
    int idx = blockIdx.x * blockDim.x + threadIdx.x;
    if (idx >= NND * HC / 4) return;
    float4 a = in[idx], b = bias[idx & 31];
    float4 r = {a.x + b.x, a.y + b.y, a.z + b.z, a.w + b.w};
    out[idx] = r;
}

// ---------- batchnorm (training stats) + LeakyReLU(0.01) ----------
__global__ void k_bn_stats(const float* __restrict__ h, float* sums, float* sumsq) {
    int ch = blockIdx.x;  // 0..127
    float s = 0.f, ss = 0.f;
    for (int i = threadIdx.x; i < NND; i += blockDim.x) {
        float v = h[(size_t)i * HC + ch];
        s += v; ss += v * v;
    }
    __shared__ float rs[256], rq[256];
    rs[threadIdx.x] = s; rq[threadIdx.x] = ss;
    __syncthreads();
    for (int o = 128; o > 0; o >>= 1) {
        if ((int)threadIdx.x < o) { rs[threadIdx.x] += rs[threadIdx.x + o]; rq[threadIdx.x] += rq[threadIdx.x + o]; }
        __syncthreads();
    }
    if (threadIdx.x == 0) { sums[ch] = rs[0]; sumsq[ch] = rq[0]; }
}
__global__ void k_bn_apply(float4* __restrict__ h, const float* sums, const float* sumsq,
                           const float* g, const float* be) {
    int idx = blockIdx.x * blockDim.x + threadIdx.x;
    if (idx >= NND * HC / 4) return;
    int cb = (idx & 31) * 4;                 // channel base
    float4 a = h[idx], r;
    float m0 = sums[cb + 0] * (1.f / NND), m1 = sums[cb + 1] * (1.f / NND);
    float m2 = sums[cb + 2] * (1.f / NND), m3 = sums[cb + 3] * (1.f / NND);
    float v;
    v = (a.x - m0) * rsqrtf(sumsq[cb + 0] * (1.f / NND) - m0 * m0 + 1e-5f) * g[cb + 0] + be[cb + 0];
    r.x = (v > 0.f) ? v : 0.01f * v;
    v = (a.y - m1) * rsqrtf(sumsq[cb + 1] * (1.f / NND) - m1 * m1 + 1e-5f) * g[cb + 1] + be[cb + 1];
    r.y = (v > 0.f) ? v : 0.01f * v;
    v = (a.z - m2) * rsqrtf(sumsq[cb + 2] * (1.f / NND) - m2 * m2 + 1e-5f) * g[cb + 2] + be[cb + 2];
    r.z = (v > 0.f) ? v : 0.01f * v;
    v = (a.w - m3) * rsqrtf(sumsq[cb + 3] * (1.f / NND) - m3 * m3 + 1e-5f) * g[cb + 3] + be[cb + 3];
    r.w = (v > 0.f) ? v : 0.01f * v;
    h[idx] = r;
}

// ---------- final linear (128->2) + softmax : one wave per node ----------
__global__ void k_final(const float* __restrict__ h, const float* __restrict__ Wf,
                        const float* __restrict__ bf, float* __restrict__ out) {
    int n = blockIdx.x * blockDim.y + threadIdx.y;
    if (n >= NND) return;
    int j = threadIdx.x;   // 0..31, owns input rows 4j..4j+3
    float4 hv  = *reinterpret_cast<const float4*>(h + (size_t)n * HC + 4 * j);
    float4 w01 = *reinterpret_cast<const float4*>(Wf + 8 * j);      // {k0c0,k0c1,k1c0,k1c1}
    float4 w23 = *reinterpret_cast<const float4*>(Wf + 8 * j + 4);  // {k2c0,k2c1,k3c0,k3c1}
    float d0 = hv.x * w01.x + hv.y * w01.z + hv.z * w23.x + hv.w * w23.z;
    float d1 = hv.x * w01.y + hv.y * w01.w + hv.z * w23.y + hv.w * w23.w;
    #pragma unroll
    for (int off = 16; off > 0; off >>= 1) {
        d0 += __shfl_xor(d0, off, 32);
        d1 += __shfl_xor(d1, off, 32);
    }
    if (j == 0) {
        d0 += bf[0]; d1 += bf[1];
        float m = fmaxf(d0, d1);
        float e0 = __expf(d0 - m), e1 = __expf(d1 - m);
        float inv = 1.f / (e0 + e1);
        out[n * 2]     = e0 * inv;
        out[n * 2 + 1] = e1 * inv;
    }
}

// ================= host orchestration =================
extern "C" void kernel_launch(void* const* d_in, const int* in_sizes, int n_in,
                              void* d_out, int out_size, void* d_ws, size_t ws_size,
                              hipStream_t stream) {
    const float* x  = (const float*)d_in[0];
    const int*   ei = (const int*)  d_in[1];
    const float* ea = (const float*)d_in[2];
    // per-layer params: Wl,bl,Wr,br,We,att,bias[,g,be]
    const float *Wl[3], *bl[3], *Wr[3], *br[3], *We[3], *att[3], *bias[3], *g[2], *be[2];
    int idx = 3;
    for (int l = 0; l < 3; ++l) {
        Wl[l]  = (const float*)d_in[idx++];
        bl[l]  = (const float*)d_in[idx++];
        Wr[l]  = (const float*)d_in[idx++];
        br[l]  = (const float*)d_in[idx++];
        We[l]  = (const float*)d_in[idx++];
        att[l] = (const float*)d_in[idx++];
        bias[l]= (const float*)d_in[idx++];
        if (l < 2) { g[l] = (const float*)d_in[idx++]; be[l] = (const float*)d_in[idx++]; }
    }
    const float* Wf = (const float*)d_in[idx++];
    const float* bf = (const float*)d_in[idx++];

    // workspace carve-up (floats; all offsets 16B-aligned)
    float* ws = (float*)d_ws;
    size_t off = 0;
    float* deg      = ws + off; off += 50176;                 // N (padded)
    float* loopattr = ws + off; off += (size_t)NND * FE;      // 800k
    float* xl       = ws + off; off += (size_t)NND * HC;      // 6.4M
    float* xr       = ws + off; off += (size_t)NND * HC;
    float* hbuf     = ws + off; off += (size_t)NND * HC;
    float* outb     = ws + off; off += (size_t)NND * HC;
    float* aex      = ws + off; off += (size_t)NET * 4;       // 6.6M
    unsigned* amax  = (unsigned*)(ws + off); off += (size_t)NND * 4;
    float* den      = ws + off; off += (size_t)NND * 4;
    float* sums     = ws + off; off += 128;
    float* sumsq    = ws + off; off += 128;

    const int T = 256;
    #define GRID1(n) dim3(((n) + T - 1) / T)

    // self-loop edge_attr = per-dst mean of incoming edge_attr
    k_zero_f32<<<GRID1(NND), T, 0, stream>>>(deg, NND);
    k_zero_f32<<<GRID1(NND * FE), T, 0, stream>>>(loopattr, NND * FE);
    k_deg_scatter<<<GRID1(NED), T, 0, stream>>>(ei, ea, deg, loopattr);
    k_loop_norm<<<GRID1(NND * FE), T, 0, stream>>>(loopattr, deg);

    dim3 gemmGrid(NND / 16, 2), gemmBlk(32, 4);
    dim3 edgeGrid((NET + 7) / 8), edgeBlk(32, 8);

    for (int l = 0; l < 3; ++l) {
        const float* in = (l == 0) ? x : hbuf;
        int K = (l == 0) ? 64 : HC;
        k_gemm_wmma<<<gemmGrid, gemmBlk, 0, stream>>>(in, Wl[l], bl[l], xl, K);
        k_gemm_wmma<<<gemmGrid, gemmBlk, 0, stream>>>(in, Wr[l], br[l], xr, K);

        k_fill_u32<<<GRID1(NND * 4), T, 0, stream>>>(amax, NND * 4, AMAX_INIT);
        k_zero_f32<<<GRID1(NND * 4), T, 0, stream>>>(den, NND * 4);
        k_zero_f32<<<GRID1(NND * HC), T, 0, stream>>>(outb, NND * HC);

        k_edge_a<<<edgeGrid, edgeBlk, 0, stream>>>(xl, xr, ei, ea, loopattr,
                                                   We[l], att[l], aex, amax);
        k_edge_b<<<GRID1(NET * 4), T, 0, stream>>>(aex, ei, amax, den);
        k_edge_c<<<edgeGrid, edgeBlk, 0, stream>>>(xl, ei, aex, den, outb);

        k_add_bias<<<GRID1(NND * HC / 4), T, 0, stream>>>((const float4*)outb,
                                                          (const float4*)bias[l],
                                                          (float4*)hbuf);
        if (l < 2) {
            k_bn_stats<<<dim3(HC), dim3(256), 0, stream>>>(hbuf, sums, sumsq);
            k_bn_apply<<<GRID1(NND * HC / 4), T, 0, stream>>>((float4*)hbuf, sums, sumsq,
                                                              g[l], be[l]);
        }
    }
    dim3 finGrid((NND + 7) / 8), finBlk(32, 8);
    k_final<<<finGrid, finBlk, 0, stream>>>(hbuf, Wf, bf, (float*)d_out);
}